// GatedSpectralStateLM_68607807586873
// MI455X (gfx1250) — compile-verified
//
#include <hip/hip_runtime.h>
#include <hip/hip_bf16.h>
#include <math.h>

#define V_SIZE 32000
#define S_SIZE 3
#define K_SIZE 4
#define B_SIZE 4
#define T_SIZE 512
#define D12    (S_SIZE * K_SIZE)   // 12
#define D24    (2 * D12)           // 24 (real || imag)
#define ROWS   (B_SIZE * T_SIZE)   // 2048

typedef float v2f __attribute__((ext_vector_type(2)));
typedef float v8f __attribute__((ext_vector_type(8)));

// ---------------------------------------------------------------------------
// Kernel 1: forward+backward gated scan (tiny, serial over T). 48 active lanes
// = (b, s, k). Gate coherence needs mean over k: 4 consecutive lanes -> shfl.
// Backward pass combines with forward states and writes the scaled S-matrix.
// ---------------------------------------------------------------------------
__device__ __forceinline__ void scan_step(int row, int t, int d, int s,
                                          const int* __restrict__ tok,
                                          const unsigned char* __restrict__ mask,
                                          const float* __restrict__ dAmp,
                                          const float* __restrict__ dPh,
                                          const float* __restrict__ gBias,
                                          float sens, float& amp, float& ph)
{
    int tk = tok[row];
    unsigned char m = mask[row];
    // torch skips the update entirely when ALL batch elements are masked at t
    unsigned char keep = mask[0 * T_SIZE + t] & mask[1 * T_SIZE + t] &
                         mask[2 * T_SIZE + t] & mask[3 * T_SIZE + t];
    float act = m ? 0.0f : 1.0f;
    float da = dAmp[tk * D12 + d] * act;
    float dp = dPh[tk * D12 + d] * act;
    float c = amp * da * __cosf(ph - dp);
    c += __shfl_xor(c, 1, 32);
    c += __shfl_xor(c, 2, 32);
    c *= 0.25f;                                     // mean over K=4
    float gate = 1.0f / (1.0f + __expf(-(gBias[tk * S_SIZE + s] + sens * c)));
    float na = gate * amp + (1.0f - gate) * da;
    float np = gate * ph  + (1.0f - gate) * dp;
    if (!keep) { amp = na; ph = np; }
}

__global__ void scan_kernel(const int* __restrict__ tok,
                            const unsigned char* __restrict__ mask,
                            const float* __restrict__ dAmp,
                            const float* __restrict__ dPh,
                            const float* __restrict__ gBias,
                            const float* __restrict__ sensP,
                            const float* __restrict__ combP,
                            const float* __restrict__ tempP,
                            float* __restrict__ Smat,   // (ROWS x 24)
                            float* __restrict__ fAmp,   // (ROWS x 12)
                            float* __restrict__ fPh)    // (ROWS x 12)
{
    const int lane = threadIdx.x;        // 0..63, active < 48
    if (lane >= B_SIZE * D12) return;
    const int b = lane / D12;
    const int d = lane % D12;            // d = s*4 + k
    const int s = d / K_SIZE;

    const float sens = sensP[0];
    const float w    = 1.0f / (1.0f + __expf(-combP[0]));   // sigmoid(combine)
    const float invT = 1.0f / (fabsf(tempP[0]) + 1e-6f);

    // ---- forward scan: emit state BEFORE update ----
    float amp = 0.0f, ph = 0.0f;
    for (int t = 0; t < T_SIZE; ++t) {
        int row = b * T_SIZE + t;
        fAmp[row * D12 + d] = amp;
        fPh [row * D12 + d] = ph;
        scan_step(row, t, d, s, tok, mask, dAmp, dPh, gBias, sens, amp, ph);
    }

    // ---- backward scan: combine with forward state, fold mask/temperature ----
    amp = 0.0f; ph = 0.0f;
    for (int t = T_SIZE - 1; t >= 0; --t) {
        int row = b * T_SIZE + t;
        float ca = w * fAmp[row * D12 + d] + (1.0f - w) * amp;
        float cp = w * fPh [row * D12 + d] + (1.0f - w) * ph;
        float scale = mask[row] ? invT : 0.0f;   // reference zeroes unmasked rows
        Smat[row * D24 + d]       = ca * __cosf(cp) * scale;
        Smat[row * D24 + D12 + d] = ca * __sinf(cp) * scale;
        scan_step(row, t, d, s, tok, mask, dAmp, dPh, gBias, sens, amp, ph);
    }
}

// ---------------------------------------------------------------------------
// Kernel 2: vocab projection matrix (32000 x 24) = [amp*cos(ph) || amp*sin(ph)]
// 3 MB -> stays L2-resident for all 128 row-tile passes of the GEMM.
// ---------------------------------------------------------------------------
__global__ void vmat_kernel(const float* __restrict__ dAmp,
                            const float* __restrict__ dPh,
                            float* __restrict__ Vmat)
{
    int i = blockIdx.x * blockDim.x + threadIdx.x;
    if (i >= V_SIZE * D12) return;
    int v = i / D12, d = i % D12;
    float a = dAmp[i], p = dPh[i];
    Vmat[v * D24 + d]       = a * __cosf(p);
    Vmat[v * D24 + D12 + d] = a * __sinf(p);
}

// ---------------------------------------------------------------------------
// Kernel 3: (2048 x 24) @ (24 x 32000) via V_WMMA_F32_16X16X4_F32.
// One wave per 16x16 tile; 8 waves/block share an m-tile (A-row reuse).
// Output is 262 MB write-once (> 192 MB L2) -> nontemporal stores.
// A layout (16x4 f32): lane&15 = M; lane>>4 selects K-pair {0,1}/{2,3}.
// B layout mirrored:   lane&15 = N; lane>>4 selects K-pair.
// C/D layout: VGPR r -> (M = r + 8*(lane>>4), N = lane&15).
// ---------------------------------------------------------------------------
__global__ __launch_bounds__(256) void logits_wmma_kernel(
        const float* __restrict__ Smat,    // (2048 x 24)
        const float* __restrict__ Vmat,    // (32000 x 24)
        float* __restrict__ out)           // (2048 x 32000)
{
    const int lane = threadIdx.x & 31;
    const int wave = threadIdx.x >> 5;
    const int m0   = blockIdx.x * 16;
    const int n0   = (blockIdx.y * 8 + wave) * 16;
    const int half = lane >> 4;            // 0 or 1
    const int l15  = lane & 15;

    const float* Srow = Smat + (size_t)(m0 + l15) * D24 + 2 * half;
    const float* Vrow = Vmat + (size_t)(n0 + l15) * D24 + 2 * half;

    v8f c = {};
#pragma unroll
    for (int kk = 0; kk < 6; ++kk) {       // K = 24 = 6 x 4
        v2f a  = *(const v2f*)(Srow + kk * 4);   // 8B-aligned (even offsets)
        v2f bb = *(const v2f*)(Vrow + kk * 4);
        c = __builtin_amdgcn_wmma_f32_16x16x4_f32(
                /*neg_a=*/false, a, /*neg_b=*/false, bb,
                /*c_mod=*/(short)0, c, /*reuse_a=*/false, /*reuse_b=*/false);
    }

    float* orow = out + (size_t)(m0 + half * 8) * V_SIZE + n0 + l15;
#pragma unroll
    for (int r = 0; r < 8; ++r)
        __builtin_nontemporal_store(c[r], orow + (size_t)r * V_SIZE);
}

// ---------------------------------------------------------------------------
extern "C" void kernel_launch(void* const* d_in, const int* in_sizes, int n_in,
                              void* d_out, int out_size, void* d_ws, size_t ws_size,
                              hipStream_t stream)
{
    (void)in_sizes; (void)n_in; (void)out_size; (void)ws_size;
    const int*           tok   = (const int*)d_in[0];
    const unsigned char* mask  = (const unsigned char*)d_in[1];   // numpy bool, 1B
    const float*         dAmp  = (const float*)d_in[2];
    const float*         dPh   = (const float*)d_in[3];
    const float*         gBias = (const float*)d_in[4];
    const float*         sens  = (const float*)d_in[5];
    const float*         comb  = (const float*)d_in[6];
    const float*         temp  = (const float*)d_in[7];
    float* out = (float*)d_out;

    float* ws   = (float*)d_ws;
    float* Smat = ws;                        // 2048*24   = 49152 f
    float* Vmat = Smat + (size_t)ROWS * D24; // 32000*24  = 768000 f
    float* fAmp = Vmat + (size_t)V_SIZE * D24;
    float* fPh  = fAmp + (size_t)ROWS * D12; // total ~3.47 MB

    hipLaunchKernelGGL(scan_kernel, dim3(1), dim3(64), 0, stream,
                       tok, mask, dAmp, dPh, gBias, sens, comb, temp,
                       Smat, fAmp, fPh);
    hipLaunchKernelGGL(vmat_kernel, dim3((V_SIZE * D12 + 255) / 256), dim3(256),
                       0, stream, dAmp, dPh, Vmat);
    hipLaunchKernelGGL(logits_wmma_kernel, dim3(ROWS / 16, V_SIZE / 16 / 8),
                       dim3(256), 0, stream, Smat, Vmat, out);
}